// VNDGCNNEncoder_88871463289219
// MI455X (gfx1250) — compile-verified
//
#include <hip/hip_runtime.h>

// VN-DGCNN encoder for gfx1250 (MI455X), wave32, fp32 WMMA (V_WMMA_F32_16X16X4_F32).
// Deterministic: no atomics; means folded in-wave (over K) or via partial sums (over N).
// Guard-free hot loops: all operands come from zero-padded staging buffers
// (xpad / hT in global, weight tiles in LDS), so the WMMA K-loops are pure
// load->wmma chains with no clamps or selects.

#define NPTS 1024
#define BATCH 8
#define KNN 20
#define CCAT 169          // 21 + 21 + 42 + 85
#define SLOPE 0.2f
#define ONE_MINUS_SLOPE 0.8f
#define VNEPS 1e-6f
#define COUT4 341
#define COUT4_PAD 352     // 22 tiles of 16
#define CIN4 169
#define WSTR 49           // LDS weight-tile row stride (odd -> conflict-free)
#define W4STR 173         // LDS final-weight row stride (odd)
#define DSTR 136          // padded feature stride: >= 3*42+6, multiple of 4 (16B rows)

typedef __attribute__((ext_vector_type(2))) float v2f;
typedef __attribute__((ext_vector_type(8))) float v8f;

static __device__ __forceinline__ v8f wmma_f32(v2f a, v2f b, v8f c) {
  return __builtin_amdgcn_wmma_f32_16x16x4_f32(false, a, false, b, (short)0, c, false, false);
}

// Load the 4x(3 xyz) B-tile slice for channels [i0, i0+4) of one point's
// contiguous hT row: 3 x b128 loads + cheap hi-selects (hi in {0,1}).
static __device__ __forceinline__ void load_b3(const float* __restrict__ row, int i0, int hi,
                                               v2f& bx, v2f& by, v2f& bz) {
  float4 w0 = *(const float4*)(row + 3 * i0);
  float4 w1 = *(const float4*)(row + 3 * i0 + 4);
  float4 w2 = *(const float4*)(row + 3 * i0 + 8);
  bx.x = hi ? w0.w : w0.x;   // e[3*hi + 0]
  by.x = hi ? w1.x : w0.y;   // e[3*hi + 1]
  bz.x = hi ? w1.y : w0.z;   // e[3*hi + 2]
  bx.y = hi ? w2.y : w1.z;   // e[6 + 3*hi + 0]
  by.y = hi ? w2.z : w1.w;   // e[6 + 3*hi + 1]
  bz.y = hi ? w2.w : w2.x;   // e[6 + 3*hi + 2]
}

// ---------------------------------------------------------------------------
// 0) Zero-padded feature copy: xpad[b][d][n], d in [0,DSTR), rows >= D are zero.
__global__ void pad_kernel(const float* __restrict__ h, int bStride, int baseOff, int D,
                           float* __restrict__ xpad) {
  int t = blockIdx.x * blockDim.x + threadIdx.x;
  if (t >= BATCH * DSTR * NPTS) return;
  int n = t & (NPTS - 1);
  int bd = t >> 10;
  int b = bd / DSTR;
  int d = bd - b * DSTR;
  float v = 0.f;
  if (d < D) v = h[(size_t)b * bStride + baseOff + (size_t)d * NPTS + n];  // wave-uniform d
  xpad[t] = v;
}

// 1) Per-point squared norms over feature dim D (zeros beyond D are free).
__global__ void norm_kernel(const float* __restrict__ xpad, int D, float* __restrict__ xx) {
  int t = blockIdx.x * blockDim.x + threadIdx.x;
  if (t >= BATCH * NPTS) return;
  int b = t >> 10, n = t & (NPTS - 1);
  const float* xp = xpad + (size_t)b * DSTR * NPTS;
  float s = 0.f;
  for (int d = 0; d < D; ++d) {
    float v = xp[(size_t)d * NPTS + n];
    s += v * v;
  }
  xx[t] = s;
}

// 2) Pairwise scores S[b,i,j] = 2<xi,xj> - |xi|^2 - |xj|^2.
//    One wave: 16 i-rows x 64 j-cols (4 WMMA tiles share each A tile).
//    Guard-free: xpad is zero beyond D, Dpad8 is a multiple of 8.
__global__ void gram_kernel(const float* __restrict__ xpad, int Dpad8,
                            const float* __restrict__ xx, float* __restrict__ S) {
  const int lane = threadIdx.x;
  const int lo = lane & 15, hi = lane >> 4;
  const int b = blockIdx.z;
  const int i0 = blockIdx.y * 16;
  const int j0 = blockIdx.x * 64;
  const float* xp = xpad + (size_t)b * DSTR * NPTS;

  v8f acc0 = {}, acc1 = {}, acc2 = {}, acc3 = {};
  for (int d0 = 0; d0 < Dpad8; d0 += 4) {
    const float* ra = xp + (size_t)(d0 + 2 * hi) * NPTS + i0 + lo;
    const float* rb = xp + (size_t)(d0 + hi) * NPTS + j0 + lo;
    __builtin_prefetch(rb + 4 * NPTS, 0, 3);
    v2f a, b0, b1, b2, b3;
    a.x  = ra[0];
    a.y  = ra[NPTS];
    b0.x = rb[0];  b0.y = rb[2 * NPTS];
    b1.x = rb[16]; b1.y = rb[2 * NPTS + 16];
    b2.x = rb[32]; b2.y = rb[2 * NPTS + 32];
    b3.x = rb[48]; b3.y = rb[2 * NPTS + 48];
    acc0 = wmma_f32(a, b0, acc0);
    acc1 = wmma_f32(a, b1, acc1);
    acc2 = wmma_f32(a, b2, acc2);
    acc3 = wmma_f32(a, b3, acc3);
  }

  float xxi[8];
#pragma unroll
  for (int r = 0; r < 8; ++r) xxi[r] = xx[b * NPTS + i0 + r + 8 * hi];
  float xxj0 = xx[b * NPTS + j0 + lo];
  float xxj1 = xx[b * NPTS + j0 + 16 + lo];
  float xxj2 = xx[b * NPTS + j0 + 32 + lo];
  float xxj3 = xx[b * NPTS + j0 + 48 + lo];
#pragma unroll
  for (int r = 0; r < 8; ++r) {
    size_t rowOff = (size_t)(b * NPTS + i0 + r + 8 * hi) * NPTS + j0 + lo;
    S[rowOff]      = 2.f * acc0[r] - xxi[r] - xxj0;
    S[rowOff + 16] = 2.f * acc1[r] - xxi[r] - xxj1;
    S[rowOff + 32] = 2.f * acc2[r] - xxi[r] - xxj2;
    S[rowOff + 48] = 2.f * acc3[r] - xxi[r] - xxj3;
  }
}

// 3) Stable top-K per row: register-resident branchless bubble insertion
//    (strict '>' keeps earlier indices first on ties, like lax.top_k).
__global__ void topk_kernel(const float* __restrict__ S, int* __restrict__ idxOut) {
  int t = blockIdx.x * blockDim.x + threadIdx.x;
  if (t >= BATCH * NPTS) return;
  const float* row = S + (size_t)t * NPTS;
  float vals[KNN];
  int inds[KNN];
#pragma unroll
  for (int k = 0; k < KNN; ++k) { vals[k] = -3.4e38f; inds[k] = 0; }
  for (int j = 0; j < NPTS; ++j) {
    float v = row[j];
    if (v > vals[KNN - 1]) {
      float cv = v;
      int ci = j;
      bool carry = false;
#pragma unroll
      for (int p = 0; p < KNN; ++p) {
        bool sw = carry || (cv > vals[p]);
        float tv = vals[p]; int ti = inds[p];
        vals[p] = sw ? cv : tv; inds[p] = sw ? ci : ti;
        cv = sw ? tv : cv;      ci = sw ? ti : ci;
        carry = sw;
      }
    }
  }
#pragma unroll
  for (int k = 0; k < KNN; ++k) idxOut[t * KNN + k] = inds[k];
}

// 4) Transpose xpad into point-major rows hT[b*N+n][d] (copies the zero pad too).
__global__ void transpose_kernel(const float* __restrict__ xpad, float* __restrict__ hT) {
  int t = blockIdx.x * blockDim.x + threadIdx.x;
  if (t >= BATCH * NPTS) return;
  int b = t >> 10, n = t & (NPTS - 1);
  const float* xp = xpad + (size_t)b * DSTR * NPTS + n;
  float* row = hT + (size_t)t * DSTR;
  for (int d = 0; d < DSTR; ++d) row[d] = xp[(size_t)d * NPTS];
}

// 5) Edge layer. Decomposition: p_k = WfA.nbr_k + (WfB - WfA).ctr, so the
//    k-invariant ctr GEMM runs once and seeds the per-neighbor accumulators.
//    6 WMMA chains (p/d x xyz); VN leaky-ReLU per lane; mean over K in-wave.
__global__ void edge_vn_kernel(const float* __restrict__ hT, int Cin, int Cout,
                               const int* __restrict__ knn,
                               const float* __restrict__ Wf, const float* __restrict__ Wd,
                               float* __restrict__ cat, int outOff) {
  __shared__ float sAf[16 * WSTR], sAd[16 * WSTR], sCf[16 * WSTR], sCd[16 * WSTR];
  const int lane = threadIdx.x;
  const int lo = lane & 15, hi = lane >> 4;
  const int b = blockIdx.z;
  const int n = blockIdx.x * 16 + lo;      // column (point) of B/D tiles
  const int o0 = blockIdx.y * 16;
  const int twoC = 2 * Cin;

  // Stage zero-padded weight tiles: WfA and (WfB - WfA), same for Wd.
  for (int idx = lane; idx < 16 * WSTR; idx += 32) {
    int m = idx / WSTR, i = idx - m * WSTR;
    int o = o0 + m;
    bool valid = (o < Cout) && (i < Cin);
    int oc = (o < Cout) ? o : (Cout - 1);
    int ic = (i < Cin) ? i : 0;
    float fa = Wf[oc * twoC + ic];
    float fb = Wf[oc * twoC + Cin + ic];
    float da = Wd[oc * twoC + ic];
    float db = Wd[oc * twoC + Cin + ic];
    sAf[idx] = valid ? fa : 0.f;
    sCf[idx] = valid ? (fb - fa) : 0.f;
    sAd[idx] = valid ? da : 0.f;
    sCd[idx] = valid ? (db - da) : 0.f;
  }
  __syncthreads();

  const float* ctrRow = hT + (size_t)(b * NPTS + n) * DSTR;

  // k-invariant center contribution.
  v8f pcx = {}, pcy = {}, pcz = {}, qcx = {}, qcy = {}, qcz = {};
  for (int i0 = 0; i0 < Cin; i0 += 4) {
    v2f af, ad, bx, by, bz;
    af.x = sCf[lo * WSTR + i0 + 2 * hi]; af.y = sCf[lo * WSTR + i0 + 2 * hi + 1];
    ad.x = sCd[lo * WSTR + i0 + 2 * hi]; ad.y = sCd[lo * WSTR + i0 + 2 * hi + 1];
    load_b3(ctrRow, i0, hi, bx, by, bz);
    pcx = wmma_f32(af, bx, pcx); pcy = wmma_f32(af, by, pcy); pcz = wmma_f32(af, bz, pcz);
    qcx = wmma_f32(ad, bx, qcx); qcy = wmma_f32(ad, by, qcy); qcz = wmma_f32(ad, bz, qcz);
  }

  float sum[8][3];
#pragma unroll
  for (int r = 0; r < 8; ++r) { sum[r][0] = 0.f; sum[r][1] = 0.f; sum[r][2] = 0.f; }

  for (int k = 0; k < KNN; ++k) {
    const int nbr = knn[(b * NPTS + n) * KNN + k];
    const float* nrow = hT + (size_t)(b * NPTS + nbr) * DSTR;
    v8f px = pcx, py = pcy, pz = pcz, qx = qcx, qy = qcy, qz = qcz;
    for (int i0 = 0; i0 < Cin; i0 += 4) {
      v2f af, ad, bx, by, bz;
      af.x = sAf[lo * WSTR + i0 + 2 * hi]; af.y = sAf[lo * WSTR + i0 + 2 * hi + 1];
      ad.x = sAd[lo * WSTR + i0 + 2 * hi]; ad.y = sAd[lo * WSTR + i0 + 2 * hi + 1];
      load_b3(nrow, i0, hi, bx, by, bz);
      px = wmma_f32(af, bx, px); py = wmma_f32(af, by, py); pz = wmma_f32(af, bz, pz);
      qx = wmma_f32(ad, bx, qx); qy = wmma_f32(ad, by, qy); qz = wmma_f32(ad, bz, qz);
    }
#pragma unroll
    for (int r = 0; r < 8; ++r) {
      float Px = px[r], Py = py[r], Pz = pz[r];
      float Dx = qx[r], Dy = qy[r], Dz = qz[r];
      float dot = Px * Dx + Py * Dy + Pz * Dz;
      float dsq = Dx * Dx + Dy * Dy + Dz * Dz;
      float coef = (dot >= 0.f) ? 0.f : dot / (dsq + VNEPS);
      sum[r][0] += SLOPE * Px + ONE_MINUS_SLOPE * (Px - coef * Dx);
      sum[r][1] += SLOPE * Py + ONE_MINUS_SLOPE * (Py - coef * Dy);
      sum[r][2] += SLOPE * Pz + ONE_MINUS_SLOPE * (Pz - coef * Dz);
    }
  }

  const float invK = 1.f / (float)KNN;
#pragma unroll
  for (int r = 0; r < 8; ++r) {
    int o = o0 + r + 8 * hi;
    if (o < Cout) {
#pragma unroll
      for (int xyz = 0; xyz < 3; ++xyz)
        cat[(size_t)((b * CCAT + outOff + o) * 3 + xyz) * NPTS + n] = sum[r][xyz] * invK;
    }
  }
}

// 6) Final-layer direction vector (Wd4 has a single output channel).
__global__ void dvec_kernel(const float* __restrict__ cat, const float* __restrict__ Wd4,
                            float* __restrict__ dv) {
  int t = blockIdx.x * blockDim.x + threadIdx.x;
  if (t >= BATCH * 3 * NPTS) return;
  int b = t / (3 * NPTS);
  int rem = t - b * 3 * NPTS;
  int xyz = rem >> 10, n = rem & (NPTS - 1);
  float s = 0.f;
  for (int i = 0; i < CIN4; ++i)
    s += Wd4[i] * cat[(size_t)((b * CCAT + i) * 3 + xyz) * NPTS + n];
  dv[t] = s;  // (b, xyz, n)
}

// 7) Final p = Wf4 @ cat (3 WMMA chains, LDS-staged zero-padded weights),
//    VN-ReLU vs dvec, deterministic partial mean over a 128-point chunk.
__global__ void finalp_kernel(const float* __restrict__ cat, const float* __restrict__ Wf4,
                              const float* __restrict__ dv, float* __restrict__ part) {
  __shared__ float sW[16 * W4STR];
  const int lane = threadIdx.x;
  const int lo = lane & 15, hi = lane >> 4;
  const int b = blockIdx.z;
  const int ot = blockIdx.y;
  const int chunk = blockIdx.x;
  const int o0 = ot * 16;

  for (int idx = lane; idx < 16 * W4STR; idx += 32) {
    int m = idx / W4STR, i = idx - m * W4STR;
    int o = o0 + m;
    bool valid = (o < COUT4) && (i < CIN4);
    int oc = (o < COUT4) ? o : (COUT4 - 1);
    int ic = (i < CIN4) ? i : 0;
    sW[idx] = valid ? Wf4[oc * CIN4 + ic] : 0.f;
  }
  __syncthreads();

  float sum[8][3];
#pragma unroll
  for (int r = 0; r < 8; ++r) { sum[r][0] = 0.f; sum[r][1] = 0.f; sum[r][2] = 0.f; }

  for (int nt = 0; nt < 8; ++nt) {
    const int n = chunk * 128 + nt * 16 + lo;
    v8f px = {}, py = {}, pz = {};
    for (int i0 = 0; i0 < CIN4; i0 += 4) {
      v2f a, bx, by, bz;
      a.x = sW[lo * W4STR + i0 + 2 * hi];
      a.y = sW[lo * W4STR + i0 + 2 * hi + 1];
      int i1 = i0 + hi, i2 = i0 + 2 + hi;            // clamp: weight side is zero-padded
      int i1c = (i1 < CIN4) ? i1 : (CIN4 - 1);
      int i2c = (i2 < CIN4) ? i2 : (CIN4 - 1);
      bx.x = cat[(size_t)((b * CCAT + i1c) * 3 + 0) * NPTS + n];
      bx.y = cat[(size_t)((b * CCAT + i2c) * 3 + 0) * NPTS + n];
      by.x = cat[(size_t)((b * CCAT + i1c) * 3 + 1) * NPTS + n];
      by.y = cat[(size_t)((b * CCAT + i2c) * 3 + 1) * NPTS + n];
      bz.x = cat[(size_t)((b * CCAT + i1c) * 3 + 2) * NPTS + n];
      bz.y = cat[(size_t)((b * CCAT + i2c) * 3 + 2) * NPTS + n];
      px = wmma_f32(a, bx, px);
      py = wmma_f32(a, by, py);
      pz = wmma_f32(a, bz, pz);
    }
    const float Dx = dv[(b * 3 + 0) * NPTS + n];
    const float Dy = dv[(b * 3 + 1) * NPTS + n];
    const float Dz = dv[(b * 3 + 2) * NPTS + n];
    const float dsq = Dx * Dx + Dy * Dy + Dz * Dz;
#pragma unroll
    for (int r = 0; r < 8; ++r) {
      float Px = px[r], Py = py[r], Pz = pz[r];
      float dot = Px * Dx + Py * Dy + Pz * Dz;
      float coef = (dot >= 0.f) ? 0.f : dot / (dsq + VNEPS);
      sum[r][0] += SLOPE * Px + ONE_MINUS_SLOPE * (Px - coef * Dx);
      sum[r][1] += SLOPE * Py + ONE_MINUS_SLOPE * (Py - coef * Dy);
      sum[r][2] += SLOPE * Pz + ONE_MINUS_SLOPE * (Pz - coef * Dz);
    }
  }

#pragma unroll
  for (int r = 0; r < 8; ++r) {
    int o = o0 + r + 8 * hi;  // < 352 always; padded region is harmless scratch
#pragma unroll
    for (int xyz = 0; xyz < 3; ++xyz)
      part[((b * COUT4_PAD + o) * 3 + xyz) * 8 + chunk] = sum[r][xyz];
  }
}

// 8) Sum the 8 chunks, divide by N -> final (8, 341, 3) output.
__global__ void reduce_kernel(const float* __restrict__ part, float* __restrict__ out) {
  int t = blockIdx.x * blockDim.x + threadIdx.x;
  if (t >= BATCH * COUT4 * 3) return;
  int b = t / (COUT4 * 3);
  int rem = t - b * COUT4 * 3;
  int o = rem / 3, xyz = rem - o * 3;
  float s = 0.f;
#pragma unroll
  for (int c = 0; c < 8; ++c) s += part[((b * COUT4_PAD + o) * 3 + xyz) * 8 + c];
  out[t] = s * (1.f / (float)NPTS);
}

// ---------------------------------------------------------------------------
extern "C" void kernel_launch(void* const* d_in, const int* in_sizes, int n_in,
                              void* d_out, int out_size, void* d_ws, size_t ws_size,
                              hipStream_t stream) {
  (void)in_sizes; (void)n_in; (void)out_size; (void)ws_size;
  const float* x = (const float*)d_in[0];
  const float* Wf[5] = {(const float*)d_in[1], (const float*)d_in[3], (const float*)d_in[5],
                        (const float*)d_in[7], (const float*)d_in[9]};
  const float* Wd[5] = {(const float*)d_in[2], (const float*)d_in[4], (const float*)d_in[6],
                        (const float*)d_in[8], (const float*)d_in[10]};
  float* out = (float*)d_out;

  // Workspace carve (~59 MB total)
  char* w = (char*)d_ws;
  float* S = (float*)w;    w += (size_t)BATCH * NPTS * NPTS * sizeof(float);     // 32 MB
  float* xx = (float*)w;   w += (size_t)BATCH * NPTS * sizeof(float);
  int* knn = (int*)w;      w += (size_t)BATCH * NPTS * KNN * sizeof(int);
  float* cat = (float*)w;  w += (size_t)BATCH * CCAT * 3 * NPTS * sizeof(float); // 16.6 MB
  float* dv = (float*)w;   w += (size_t)BATCH * 3 * NPTS * sizeof(float);
  float* part = (float*)w; w += (size_t)BATCH * COUT4_PAD * 3 * 8 * sizeof(float);
  float* hT = (float*)w;   w += (size_t)BATCH * NPTS * DSTR * sizeof(float);     // 4.5 MB
  float* xpad = (float*)w; // BATCH * DSTR * NPTS floats (4.5 MB)

  struct Layer { const float* h; int bStride, baseOff, Cin, Cout, outOff; };
  const Layer L[4] = {
      {x,   3 * NPTS,        0,             1,  21, 0},
      {cat, CCAT * 3 * NPTS, 0 * 3 * NPTS,  21, 21, 21},
      {cat, CCAT * 3 * NPTS, 21 * 3 * NPTS, 21, 42, 42},
      {cat, CCAT * 3 * NPTS, 42 * 3 * NPTS, 42, 85, 84},
  };

  for (int l = 0; l < 4; ++l) {
    const int D = 3 * L[l].Cin;
    const int Dpad8 = (D + 7) & ~7;
    pad_kernel<<<(BATCH * DSTR * NPTS + 255) / 256, 256, 0, stream>>>(
        L[l].h, L[l].bStride, L[l].baseOff, D, xpad);
    norm_kernel<<<(BATCH * NPTS + 255) / 256, 256, 0, stream>>>(xpad, D, xx);
    gram_kernel<<<dim3(NPTS / 64, NPTS / 16, BATCH), 32, 0, stream>>>(xpad, Dpad8, xx, S);
    topk_kernel<<<(BATCH * NPTS + 63) / 64, 64, 0, stream>>>(S, knn);
    transpose_kernel<<<(BATCH * NPTS + 255) / 256, 256, 0, stream>>>(xpad, hT);
    edge_vn_kernel<<<dim3(NPTS / 16, (L[l].Cout + 15) / 16, BATCH), 32, 0, stream>>>(
        hT, L[l].Cin, L[l].Cout, knn, Wf[l], Wd[l], cat, L[l].outOff);
  }

  dvec_kernel<<<(BATCH * 3 * NPTS + 255) / 256, 256, 0, stream>>>(cat, Wd[4], dv);
  finalp_kernel<<<dim3(8, COUT4_PAD / 16, BATCH), 32, 0, stream>>>(cat, Wf[4], dv, part);
  reduce_kernel<<<(BATCH * COUT4 * 3 + 255) / 256, 256, 0, stream>>>(part, out);
}